// Article2Graph_11630771437813
// MI455X (gfx1250) — compile-verified
//
#include <hip/hip_runtime.h>

#define NN    8192
#define FEAT  256
#define SLOPE 0.01f
#define NEGM  (-1e30f)

typedef __attribute__((ext_vector_type(2))) float v2f;
typedef __attribute__((ext_vector_type(4))) float v4f;
typedef __attribute__((ext_vector_type(8))) float v8f;

__device__ __forceinline__ float lrelu(float z) { return z > 0.f ? z : SLOPE * z; }

// ---------------------------------------------------------------------------
// C[M,FEAT] = A[M,FEAT] @ W[FEAT,FEAT] via V_WMMA_F32_16X16X4_F32 (wave32).
// Block = 8 waves = one 16-row stripe (mt = blockIdx.x). Wave w computes the
// two 16x16 tiles nt = w and nt = w+8, reusing its A fragment for 2 WMMAs.
// Layouts (CDNA5 ISA):
//   A 16x4 : lane l -> row = l&15, vgpr v holds K = 2*(l>>4)+v
//   B 4x16 : lane l -> col = l&15, vgpr v holds K = 2*(l>>4)+v
//   C 16x16: lane l -> col = l&15, vgpr v holds row = v + 8*(l>>4)
// GATHER=true fuses the embedding lookup (layer 1: A row = emb[idx[row]]).
// ---------------------------------------------------------------------------
template <bool GATHER>
__global__ __launch_bounds__(256)
void gemm_xw_wmma(const float* __restrict__ A, const int* __restrict__ idx,
                  const float* __restrict__ W, float* __restrict__ C) {
    const int lane = threadIdx.x & 31;
    const int wave = threadIdx.x >> 5;
    const int mt   = blockIdx.x;              // NN/16 = 512 row tiles
    const int hi   = lane >> 4;               // lane half (0|1)
    const int l15  = lane & 15;
    const int row  = mt * 16 + l15;
    const int col0 = wave * 16 + l15;         // nt = wave
    const int col1 = col0 + 128;              // nt = wave + 8
    const float* arow = GATHER ? (A + (size_t)idx[row] * FEAT)
                               : (A + (size_t)row * FEAT);
    const int koff = 2 * hi;

    v8f acc0 = {}, acc1 = {};
#pragma unroll 4
    for (int k = 0; k < FEAT; k += 4) {
        v2f a = *(const v2f*)(arow + k + koff);       // 8B aligned
        const float* wr0 = W + (size_t)(k + koff) * FEAT;
        const float* wr1 = W + (size_t)(k + koff + 1) * FEAT;
        v2f b0; b0.x = wr0[col0]; b0.y = wr1[col0];
        v2f b1; b1.x = wr0[col1]; b1.y = wr1[col1];
        acc0 = __builtin_amdgcn_wmma_f32_16x16x4_f32(false, a, false, b0,
                                                     (short)0, acc0, false, false);
        acc1 = __builtin_amdgcn_wmma_f32_16x16x4_f32(false, a, false, b1,
                                                     (short)0, acc1, false, false);
    }
    float* crow = C + (size_t)(mt * 16 + 8 * hi) * FEAT;
#pragma unroll
    for (int v = 0; v < 8; ++v) {
        crow[(size_t)v * FEAT + col0] = acc0[v];
        crow[(size_t)v * FEAT + col1] = acc1[v];
    }
}

// ---------------------------------------------------------------------------
// s[i] = h[i,:] . a_src ; d[i] = h[i,:] . a_dst   (one wave32 per row)
// ---------------------------------------------------------------------------
__global__ __launch_bounds__(256)
void sd_kernel(const float* __restrict__ h, const float* __restrict__ a_src,
               const float* __restrict__ a_dst, float* __restrict__ s,
               float* __restrict__ d) {
    const int lane = threadIdx.x & 31;
    const int wave = threadIdx.x >> 5;
    const int row  = blockIdx.x * 8 + wave;
    const float* hr = h + (size_t)row * FEAT;
    float ss = 0.f, dd = 0.f;
    for (int k = lane; k < FEAT; k += 32) {
        float v = hr[k];
        ss = fmaf(v, a_src[k], ss);
        dd = fmaf(v, a_dst[k], dd);
    }
    for (int off = 16; off > 0; off >>= 1) {
        ss += __shfl_xor(ss, off, 32);
        dd += __shfl_xor(dd, off, 32);
    }
    if (lane == 0) { s[row] = ss; d[row] = dd; }
}

// ---------------------------------------------------------------------------
// Fused masked-softmax + attention write + sparse aggregation, one row per
// 256-thread block:
//   scan A: stream adj row, online (max,sum) with exp only on ~33 unmasked
//           entries; block tree-reduce (m,l).
//   scan B: re-read the row (L2-resident: working set = active rows * 32KB),
//           write normalized attn (non-temporal), compact nonzeros with a
//           deterministic prefix scan, accumulate out[i,f] in a register
//           (thread tid owns feature tid), add optional residual.
// ---------------------------------------------------------------------------
__global__ __launch_bounds__(256)
void attn_fused(const float* __restrict__ adj, const float* __restrict__ s,
                const float* __restrict__ d, const float* __restrict__ h,
                const float* __restrict__ resid, float* __restrict__ attnOut,
                float* __restrict__ out) {
    __shared__ float dsh[NN];
    __shared__ float rm[256], rl[256];
    __shared__ int   cnts[256];
    __shared__ int   pj[1024];
    __shared__ float pa[1024];
    const int i = blockIdx.x, tid = threadIdx.x;
    for (int k = tid; k < NN; k += 256) dsh[k] = d[k];
    __syncthreads();

    const float si = s[i];
    const float* arow = adj + (size_t)i * NN;

    // ---- scan A: online (m, l) over unmasked entries ----
    float m = NEGM, l = 0.f;
    for (int base = 0; base < NN; base += 1024) {
        const int j0 = base + tid * 4;
        if (base + 2048 <= NN) __builtin_prefetch(arow + j0 + 1024, 0, 0);
        v4f a4 = *(const v4f*)(arow + j0);
#pragma unroll
        for (int q = 0; q < 4; ++q) {
            if (a4[q] > 0.f) {
                float e = lrelu(si + dsh[j0 + q]);
                if (e > m) { l = l * __expf(m - e) + 1.f; m = e; }
                else       { l += __expf(e - m); }
            }
        }
    }
    rm[tid] = m; rl[tid] = l;
    __syncthreads();
    for (int off = 128; off > 0; off >>= 1) {
        if (tid < off) {
            float m2 = rm[tid + off], l2 = rl[tid + off];
            float M = fmaxf(m, m2);
            l = l * __expf(m - M) + l2 * __expf(m2 - M);
            m = M;
            rm[tid] = m; rl[tid] = l;
        }
        __syncthreads();
    }
    const float mi   = rm[0];
    const float invl = 1.f / rl[0];

    // ---- scan B: normalized attn + deterministic sparse aggregation ----
    float* wrow = attnOut + (size_t)i * NN;
    float acc = 0.f;
    for (int base = 0; base < NN; base += 1024) {
        const int j0 = base + tid * 4;
        v4f a4 = *(const v4f*)(arow + j0);           // L2 hit (just streamed)
        v4f w4;
        int msk = 0, cnt = 0;
#pragma unroll
        for (int q = 0; q < 4; ++q) {
            float attn = 0.f;
            if (a4[q] > 0.f) {
                attn = __expf(lrelu(si + dsh[j0 + q]) - mi) * invl;
                msk |= (1 << q);
                ++cnt;
            }
            w4[q] = attn;
        }
        __builtin_nontemporal_store(w4, (v4f*)(wrow + j0));  // write-once data

        // deterministic compaction: inclusive Hillis-Steele scan of counts
        cnts[tid] = cnt;
        __syncthreads();
        for (int off = 1; off < 256; off <<= 1) {
            int v   = cnts[tid];
            int add = (tid >= off) ? cnts[tid - off] : 0;
            __syncthreads();
            cnts[tid] = v + add;
            __syncthreads();
        }
        const int total = cnts[255];
        int ofs = cnts[tid] - cnt;   // exclusive offset
        if (msk) {
#pragma unroll
            for (int q = 0; q < 4; ++q)
                if (msk & (1 << q)) { pj[ofs] = j0 + q; pa[ofs] = w4[q]; ++ofs; }
        }
        __syncthreads();
        for (int p = 0; p < total; ++p)                  // ~4 pairs/chunk avg
            acc = fmaf(pa[p], h[(size_t)pj[p] * FEAT + tid], acc);
        __syncthreads();
    }
    if (resid) acc += resid[(size_t)i * FEAT + tid];
    out[(size_t)i * FEAT + tid] = acc;
}

// ---------------------------------------------------------------------------
// docMean = mean over rows of document, two-stage deterministic reduce
// ---------------------------------------------------------------------------
__global__ __launch_bounds__(256)
void mean_partial(const float* __restrict__ doc, float* __restrict__ part) {
    const int b = blockIdx.x, tid = threadIdx.x;
    const int rows = NN / 64;
    float acc = 0.f;
    const int r0 = b * rows;
    for (int r = r0; r < r0 + rows; ++r)
        acc += doc[(size_t)r * FEAT + tid];
    part[b * FEAT + tid] = acc;
}

__global__ __launch_bounds__(256)
void mean_final(const float* __restrict__ part, float* __restrict__ docMean) {
    const int tid = threadIdx.x;
    float acc = 0.f;
    for (int b = 0; b < 64; ++b) acc += part[b * FEAT + tid];
    docMean[tid] = acc * (1.f / NN);
}

// ---------------------------------------------------------------------------
extern "C" void kernel_launch(void* const* d_in, const int* in_sizes, int n_in,
                              void* d_out, int out_size, void* d_ws, size_t ws_size,
                              hipStream_t stream) {
    (void)in_sizes; (void)n_in; (void)out_size; (void)ws_size;
    const int*   inDoc = (const int*)d_in[0];
    const float* adj0  = (const float*)d_in[1];
    const float* adj1  = (const float*)d_in[2];
    const float* emb   = (const float*)d_in[3];
    const float* W1    = (const float*)d_in[4];
    const float* a1s   = (const float*)d_in[5];
    const float* a1d   = (const float*)d_in[6];
    const float* W2    = (const float*)d_in[7];
    const float* a2s   = (const float*)d_in[8];
    const float* a2d   = (const float*)d_in[9];

    float* outp    = (float*)d_out;
    float* docMean = outp;
    float* sattn   = outp + FEAT;
    float* dattn   = outp + FEAT + (size_t)NN * NN;

    char* ws = (char*)d_ws;
    const size_t matB = (size_t)NN * FEAT * sizeof(float);   // 8 MB
    const size_t vecB = (size_t)NN * sizeof(float);          // 32 KB
    float* h1   = (float*)ws;  ws += matB;
    float* out1 = (float*)ws;  ws += matB;   // layer-1 output ("words")
    float* h2   = (float*)ws;  ws += matB;
    float* doc  = (float*)ws;  ws += matB;   // layer-2 output + residual
    float* s1 = (float*)ws; ws += vecB;  float* d1 = (float*)ws; ws += vecB;
    float* s2 = (float*)ws; ws += vecB;  float* d2 = (float*)ws; ws += vecB;
    float* part = (float*)ws;  ws += 64 * FEAT * sizeof(float);

    // ---- Layer 1 ----
    gemm_xw_wmma<true><<<NN / 16, 256, 0, stream>>>(emb, inDoc, W1, h1);
    sd_kernel<<<NN / 8, 256, 0, stream>>>(h1, a1s, a1d, s1, d1);
    attn_fused<<<NN, 256, 0, stream>>>(adj0, s1, d1, h1, nullptr, sattn, out1);

    // ---- Layer 2 (residual folded in) ----
    gemm_xw_wmma<false><<<NN / 16, 256, 0, stream>>>(out1, nullptr, W2, h2);
    sd_kernel<<<NN / 8, 256, 0, stream>>>(h2, a2s, a2d, s2, d2);
    attn_fused<<<NN, 256, 0, stream>>>(adj1, s2, d2, h2, out1, dattn, doc);

    // ---- Mean ----
    mean_partial<<<64, 256, 0, stream>>>(doc, part);
    mean_final<<<1, 256, 0, stream>>>(part, docMean);
}